// SimpleRetention_40372692583282
// MI455X (gfx1250) — compile-verified
//
#include <hip/hip_runtime.h>
#include <hip/hip_bf16.h>
#include <stdint.h>

typedef __attribute__((ext_vector_type(2))) float v2f;
typedef __attribute__((ext_vector_type(8))) float v8f;

#define BATCH 8
#define CLEN  2048
#define DDIM  256
#define HDIM  64

// log2(0.9)
constexpr float kL2K = -0.15200309344504997f;
// -ln(10000)/256
constexpr float kPE = -0.03597789207803197f;

static __device__ __forceinline__ v8f wmma4(v2f a, v2f b, v8f c) {
  // D = A(16x4,f32) * B(4x16,f32) + C(16x16,f32)
  return __builtin_amdgcn_wmma_f32_16x16x4_f32(
      /*neg_a=*/false, a, /*neg_b=*/false, b,
      /*c_mod=*/(short)0, c, /*reuse_a=*/false, /*reuse_b=*/false);
}

// CDNA5 async copy: global -> LDS, 16B per lane, tracked with ASYNCcnt.
static __device__ __forceinline__ void async_load_b128(uint32_t lds_off,
                                                       const float* gptr) {
  asm volatile("global_load_async_to_lds_b128 %0, %1, off"
               :
               : "v"(lds_off), "v"((uint64_t)(uintptr_t)gptr)
               : "memory");
}
static __device__ __forceinline__ void wait_async0() {
  asm volatile("s_wait_asynccnt 0" ::: "memory");
}
static __device__ __forceinline__ void wait_ds0() {
  asm volatile("s_wait_dscnt 0" ::: "memory");
}

// ---------------------------------------------------------------------------
// Kernel 1: per-batch scan over dones -> cs (episode count), xi, w_last, delta
// ---------------------------------------------------------------------------
__global__ __launch_bounds__(32) void scan_kernel(
    const unsigned char* __restrict__ dones,
    float* __restrict__ cs, float* __restrict__ xi,
    float* __restrict__ wlast, float* __restrict__ delta) {
  const int b = blockIdx.x;
  const int lane = threadIdx.x;
  __shared__ float s_cs[CLEN];
  if (lane == 0) {
    float run = 0.0f, surv = 1.0f, d1 = 1.0f;
    for (int c = 0; c < CLEN; ++c) {
      const float df = dones[b * CLEN + c] ? 1.0f : 0.0f;
      run += df;
      surv *= (1.0f - df);
      s_cs[c] = run;
      cs[b * CLEN + c] = run;
      xi[b * CLEN + c] = __builtin_amdgcn_exp2f((float)c * kL2K) * surv;
      if (((c & 3) == 0) && (df != 0.0f)) d1 = 0.0f;  // chunk starts: c % 4 == 0
    }
    delta[b] = d1;
  }
  __syncthreads();
  const float cl = s_cs[CLEN - 1];
  for (int c = lane; c < CLEN; c += 32) {
    const float w = (s_cs[c] == cl)
                        ? __builtin_amdgcn_exp2f((float)(CLEN - 1 - c) * kL2K)
                        : 0.0f;
    wlast[b * CLEN + c] = w;
  }
}

// ---------------------------------------------------------------------------
// Kernel 2: fused positional encoding + projections  q/k/v = (X + pe) @ W
// One block = (ctile, batch, matrix); 4 waves, each owns one 16-col tile of H.
// K-sliced LDS staging (A-tile + W^T) so WMMA fragments are b64 LDS reads.
// ---------------------------------------------------------------------------
__global__ __launch_bounds__(128) void proj_kernel(
    const float* __restrict__ query, const float* __restrict__ key,
    const float* __restrict__ value, const int* __restrict__ step_count,
    const float* __restrict__ wq, const float* __restrict__ wk,
    const float* __restrict__ wv, float* __restrict__ q, float* __restrict__ k,
    float* __restrict__ v) {
  constexpr int KS = 64;
  __shared__ float sA[16][KS];     // 4 KB: input rows with PE added
  __shared__ float sWT[HDIM][KS];  // 16 KB: W^T slice (contiguous along K)
  const int ct = blockIdx.x, b = blockIdx.y, mat = blockIdx.z;
  const float* X = (mat == 0) ? query : (mat == 1 ? key : value);
  const float* W = (mat == 0) ? wq : (mat == 1 ? wk : wv);
  float* O = (mat == 0) ? q : (mat == 1 ? k : v);
  const int tid = threadIdx.x;
  const int c0 = ct * 16;
  const int wave = tid >> 5, lane = tid & 31;
  const int half = lane >> 4, nl = lane & 15;
  const int n0 = wave * 16;
  v8f acc = {0.f, 0.f, 0.f, 0.f, 0.f, 0.f, 0.f, 0.f};
  for (int ks = 0; ks < DDIM; ks += KS) {
    // stage A slice (+ positional encoding)
    for (int e = tid; e < 16 * KS; e += 128) {
      const int m = e / KS, dd = e % KS, d = ks + dd;
      const float pos = (float)step_count[b * CLEN + c0 + m];
      const float x = pos * __expf((float)(d & ~1) * kPE);
      const float pe = (d & 1) ? __cosf(x) : __sinf(x);
      sA[m][dd] = X[((size_t)(b * CLEN + c0 + m)) * DDIM + d] + pe;
    }
    // stage W^T slice
    for (int e = tid; e < HDIM * KS; e += 128) {
      const int n = e / KS, dd = e % KS;
      sWT[n][dd] = W[(size_t)(ks + dd) * HDIM + n];
    }
    __syncthreads();
    for (int kk = 0; kk < KS; kk += 4) {
      const int ko = kk + half * 2;
      v2f a, bb;
      a.x = sA[nl][ko];           // A[m][k], A[m][k+1]
      a.y = sA[nl][ko + 1];
      bb.x = sWT[n0 + nl][ko];    // B[k][n], B[k+1][n]  (from W^T)
      bb.y = sWT[n0 + nl][ko + 1];
      acc = wmma4(a, bb, acc);
    }
    __syncthreads();
  }
  for (int p = 0; p < 8; ++p)
    O[((size_t)(b * CLEN + c0 + p + half * 8)) * HDIM + n0 + nl] = acc[p];
}

// ---------------------------------------------------------------------------
// Kernel 3: main retention.  One wave per 16-row i-tile.
// q/k/v 16x64 tiles are contiguous 4KB blocks -> staged into LDS with
// global_load_async_to_lds_b128 (ASYNCcnt), WMMA fragments read from LDS except hstate (global).
// acc = (q @ hstate) * xi  (cross-chunk), then for each j-tile <= i-tile:
//   s = (q @ k^T) * decay(i,j);  acc += s @ v    (s via 1KB LDS relayout)
// ---------------------------------------------------------------------------
__global__ __launch_bounds__(32) void retention_main(
    const float* __restrict__ q, const float* __restrict__ k,
    const float* __restrict__ v, const float* __restrict__ hstate,
    const float* __restrict__ cs, const float* __restrict__ xi,
    float* __restrict__ ret) {
  __shared__ float sQ[16 * HDIM];  // 4 KB, staged once per i-tile
  __shared__ float sK[16 * HDIM];  // 4 KB, refilled per j-tile
  __shared__ float sV[16 * HDIM];  // 4 KB, refilled per j-tile
  __shared__ float s_sc[16 * 16];  // 1 KB score relayout (C-layout -> A-frags)
  const int it = blockIdx.x, b = blockIdx.y;
  const int lane = threadIdx.x;
  const int half = lane >> 4, nl = lane & 15;
  const int i0 = it * 16;

  const uint32_t qb = (uint32_t)(uintptr_t)(void*)sQ;
  const uint32_t kbase = (uint32_t)(uintptr_t)(void*)sK;
  const uint32_t vbase = (uint32_t)(uintptr_t)(void*)sV;

  // stage q tile: contiguous 4KB, 8 coalesced async b128 per wave
  {
    const float* qg = q + ((size_t)(b * CLEN) + i0) * HDIM;
    for (int r = 0; r < 8; ++r)
      async_load_b128(qb + (uint32_t)((r * 32 + lane) * 16),
                      qg + (r * 32 + lane) * 4);
    wait_async0();
  }

  // --- cross chunk: acc[nt] = q_tile @ hstate, then scale rows by xi ---
  v8f acc[4];
  for (int nt = 0; nt < 4; ++nt) {
    v8f c = {0.f, 0.f, 0.f, 0.f, 0.f, 0.f, 0.f, 0.f};
    for (int kk = 0; kk < HDIM; kk += 4) {
      const int ko = kk + half * 2;
      v2f a, bb;
      a.x = sQ[nl * HDIM + ko];
      a.y = sQ[nl * HDIM + ko + 1];
      const float* hp = hstate + ((size_t)(b * HDIM + ko)) * HDIM + nt * 16 + nl;
      bb.x = hp[0];
      bb.y = hp[HDIM];
      c = wmma4(a, bb, c);
    }
    acc[nt] = c;
  }
  // per-lane loop invariants: xi row scale, episode counts, decay factors
  float cs_i[8], krow[8];
  const float kinv = __builtin_amdgcn_exp2f(-(float)nl * kL2K);  // kappa^-nl
  for (int p = 0; p < 8; ++p) {
    const int i = i0 + p + half * 8;
    const float xr = xi[b * CLEN + i];
    cs_i[p] = cs[b * CLEN + i];
    krow[p] = __builtin_amdgcn_exp2f((float)(p + half * 8) * kL2K) * kinv;
    for (int nt = 0; nt < 4; ++nt) acc[nt][p] *= xr;
  }

  // --- inner chunk over lower-triangular j-tiles ---
  for (int jt = 0; jt <= it; ++jt) {
    const int j0 = jt * 16;
    // async LDS writes are NOT ordered with this wave's DS reads: make sure
    // previous tile's DS reads retired before overwriting sK/sV.
    wait_ds0();
    const float* kg = k + ((size_t)(b * CLEN) + j0) * HDIM;
    const float* vg = v + ((size_t)(b * CLEN) + j0) * HDIM;
    for (int r = 0; r < 8; ++r) {
      async_load_b128(kbase + (uint32_t)((r * 32 + lane) * 16),
                      kg + (r * 32 + lane) * 4);
      async_load_b128(vbase + (uint32_t)((r * 32 + lane) * 16),
                      vg + (r * 32 + lane) * 4);
    }
    wait_async0();

    // scores = q_tile(16xH) @ k_tile^T(Hx16); B[kc][n] = sK[n*H + kc]
    v8f s = {0.f, 0.f, 0.f, 0.f, 0.f, 0.f, 0.f, 0.f};
    for (int kk = 0; kk < HDIM; kk += 4) {
      const int ko = kk + half * 2;
      v2f a, bb;
      a.x = sQ[nl * HDIM + ko];
      a.y = sQ[nl * HDIM + ko + 1];
      bb.x = sK[nl * HDIM + ko];
      bb.y = sK[nl * HDIM + ko + 1];
      s = wmma4(a, bb, s);
    }
    // decay mask: kappa^(i-j) = kappa^(m-nl) * kappa^(i0-j0), same episode
    const float cs_j = cs[b * CLEN + j0 + nl];
    const float ktile = __builtin_amdgcn_exp2f((float)(i0 - j0) * kL2K);
    for (int p = 0; p < 8; ++p) {
      const int i = i0 + p + half * 8;
      const int j = j0 + nl;
      const float d =
          (i >= j && cs_i[p] == cs_j) ? (krow[p] * ktile) : 0.0f;
      s[p] *= d;
    }
    // relayout scores C-layout -> A-fragments through LDS
    for (int p = 0; p < 8; ++p) s_sc[(p + half * 8) * 16 + nl] = s[p];
    __syncthreads();  // single-wave block: S_NOP barrier + ds wait
    v2f afr[4];
    for (int k2 = 0; k2 < 4; ++k2) {
      const int ko = k2 * 4 + half * 2;
      afr[k2].x = s_sc[nl * 16 + ko];
      afr[k2].y = s_sc[nl * 16 + ko + 1];
    }
    __syncthreads();
    // acc += scores(16x16) @ v_tile(16xH); B[kc][n] = sV[kc*H + n]
    for (int nt = 0; nt < 4; ++nt) {
      for (int k2 = 0; k2 < 4; ++k2) {
        const int ko = k2 * 4 + half * 2;
        v2f bb;
        bb.x = sV[ko * HDIM + nt * 16 + nl];
        bb.y = sV[(ko + 1) * HDIM + nt * 16 + nl];
        acc[nt] = wmma4(afr[k2], bb, acc[nt]);
      }
    }
  }
  for (int nt = 0; nt < 4; ++nt)
    for (int p = 0; p < 8; ++p)
      ret[((size_t)(b * CLEN + i0 + p + half * 8)) * HDIM + nt * 16 + nl] =
          acc[nt][p];
}

// ---------------------------------------------------------------------------
// Kernel 4: next_hstate = k^T @ (v * w_last) + hstate * kappa^512 * delta
// One wave per 16x16 output tile, K = CLEN chained fp32 WMMA.
// ---------------------------------------------------------------------------
__global__ __launch_bounds__(32) void hstate_kernel(
    const float* __restrict__ k, const float* __restrict__ v,
    const float* __restrict__ wlast, const float* __restrict__ hstate,
    const float* __restrict__ delta, float* __restrict__ out_h) {
  const int tile = blockIdx.x, b = blockIdx.y;
  const int mt = tile >> 2, nt = tile & 3;
  const int lane = threadIdx.x;
  const int half = lane >> 4, nl = lane & 15;
  v8f c = {0.f, 0.f, 0.f, 0.f, 0.f, 0.f, 0.f, 0.f};
  for (int kk = 0; kk < CLEN; kk += 4) {
    const int ko = kk + half * 2;
    v2f a, bb;
    // A[m][c] = k[b, c, mt*16+m]
    a.x = k[((size_t)(b * CLEN + ko)) * HDIM + mt * 16 + nl];
    a.y = k[((size_t)(b * CLEN + ko + 1)) * HDIM + mt * 16 + nl];
    const float wl0 = wlast[b * CLEN + ko];
    const float wl1 = wlast[b * CLEN + ko + 1];
    bb.x = v[((size_t)(b * CLEN + ko)) * HDIM + nt * 16 + nl] * wl0;
    bb.y = v[((size_t)(b * CLEN + ko + 1)) * HDIM + nt * 16 + nl] * wl1;
    c = wmma4(a, bb, c);
  }
  const float cd =
      __builtin_amdgcn_exp2f(512.0f * kL2K) * delta[b];  // kappa^(C/4)
  for (int p = 0; p < 8; ++p) {
    const int m = mt * 16 + p + half * 8;
    const int n = nt * 16 + nl;
    const size_t idx = ((size_t)(b * HDIM + m)) * HDIM + n;
    out_h[idx] = c[p] + hstate[idx] * cd;
  }
}

// ---------------------------------------------------------------------------
extern "C" void kernel_launch(void* const* d_in, const int* in_sizes, int n_in,
                              void* d_out, int out_size, void* d_ws,
                              size_t ws_size, hipStream_t stream) {
  const float* key = (const float*)d_in[0];
  const float* query = (const float*)d_in[1];
  const float* value = (const float*)d_in[2];
  const float* hstate = (const float*)d_in[3];
  const unsigned char* dones = (const unsigned char*)d_in[4];
  const int* step_count = (const int*)d_in[5];
  const float* wq = (const float*)d_in[6];
  const float* wk = (const float*)d_in[7];
  const float* wv = (const float*)d_in[8];

  float* ret = (float*)d_out;                          // (B,C,H)
  float* next_h = ret + (size_t)BATCH * CLEN * HDIM;   // (B,H,H)

  const size_t bch = (size_t)BATCH * CLEN * HDIM;
  float* q = (float*)d_ws;
  float* k = q + bch;
  float* v = k + bch;
  float* cs = v + bch;
  float* xi = cs + BATCH * CLEN;
  float* wl = xi + BATCH * CLEN;
  float* dl = wl + BATCH * CLEN;

  scan_kernel<<<dim3(BATCH), dim3(32), 0, stream>>>(dones, cs, xi, wl, dl);
  proj_kernel<<<dim3(CLEN / 16, BATCH, 3), dim3(128), 0, stream>>>(
      query, key, value, step_count, wq, wk, wv, q, k, v);
  retention_main<<<dim3(CLEN / 16, BATCH), dim3(32), 0, stream>>>(
      q, k, v, hstate, cs, xi, ret);
  hstate_kernel<<<dim3(16, BATCH), dim3(32), 0, stream>>>(k, v, wl, hstate, dl,
                                                          next_h);
}